// LogEig_12584254177380
// MI455X (gfx1250) — compile-verified
//
#include <hip/hip_runtime.h>
#include <hip/hip_bf16.h>
#include <math.h>

// ---------------------------------------------------------------------------
// LogEig of 32768 independent 32x32 SPD matrices via inverse scaling-and-
// squaring (matmul-only), mapped onto V_WMMA_F32_16X16X4_F32 + TDM.
//
//   A = P / trace(P)                      (eigs in (0,1], ||I-A||<1)
//   repeat m=6: A <- sqrt(A)              (coupled Newton-Schulz, matmul-only)
//   E = A - I
//   log P = log(c)*I + 2^m * (E - E^2/2 + E^3/3 - E^4/4)
//
// NS iteration is fused into the WMMAs:
//   W' = -0.5*(Z*Y)            (scaled at accumulator writeback)
//   Y' = Y*W' + 1.5*Y          (C operand preloaded with 1.5*Y tile)
//   Z' = W'*Z + 1.5*Z          (C operand preloaded with 1.5*Z tile)
// Iterations run in ping-pong pairs (Y,Z)<->(NY,NZ): no copy pass needed.
//
// Input tile is DMA'd into LDS by the Tensor Data Mover with hardware row
// padding (pad 1 DWORD every 32 DWORDs -> 33-float pitch, bank-conflict-free
// K-strided A-operand loads).
// ---------------------------------------------------------------------------

#define LDW 33  // LDS row pitch in floats (32 + 1 pad), matches TDM pad cfg

typedef float        v2f __attribute__((ext_vector_type(2)));
typedef float        v8f __attribute__((ext_vector_type(8)));
typedef unsigned int v4u __attribute__((ext_vector_type(4)));
typedef int          v8i __attribute__((ext_vector_type(8)));
typedef int          v4i __attribute__((ext_vector_type(4)));

// D = dscale * (A * B) for 32x32 f32 matrices in LDS (row pitch LDW).
// Each of the 4 waves computes tile (ti,tj); 8 K-steps of K=4 per tile.
// VGPR layouts per CDNA5 ISA 7.12.2.
__device__ __forceinline__ void mm32s(const float* __restrict__ A,
                                      const float* __restrict__ B,
                                      float* __restrict__ D, float dscale,
                                      int lane, int ti, int tj) {
  const int half = lane >> 4;  // selects K+2 (A,B) and M+8 (C/D)
  const int l    = lane & 15;
  v8f acc = {0.f, 0.f, 0.f, 0.f, 0.f, 0.f, 0.f, 0.f};
#pragma unroll
  for (int kk = 0; kk < 8; ++kk) {
    const int k0 = kk * 4 + half * 2;
    v2f a, b;
    a.x = A[(ti * 16 + l) * LDW + k0];
    a.y = A[(ti * 16 + l) * LDW + k0 + 1];
    b.x = B[(k0)     * LDW + tj * 16 + l];
    b.y = B[(k0 + 1) * LDW + tj * 16 + l];
    acc = __builtin_amdgcn_wmma_f32_16x16x4_f32(false, a, false, b,
                                                (short)0, acc, false, false);
  }
#pragma unroll
  for (int v = 0; v < 8; ++v) {
    const int row = ti * 16 + half * 8 + v;
    D[row * LDW + tj * 16 + l] = dscale * acc[v];
  }
}

// D = A * B + cscale * C  (C operand preloaded into the WMMA accumulator).
__device__ __forceinline__ void mm32c(const float* __restrict__ A,
                                      const float* __restrict__ B,
                                      const float* __restrict__ Cm,
                                      float cscale, float* __restrict__ D,
                                      int lane, int ti, int tj) {
  const int half = lane >> 4;
  const int l    = lane & 15;
  v8f acc;
#pragma unroll
  for (int v = 0; v < 8; ++v) {
    const int row = ti * 16 + half * 8 + v;
    acc[v] = cscale * Cm[row * LDW + tj * 16 + l];
  }
#pragma unroll
  for (int kk = 0; kk < 8; ++kk) {
    const int k0 = kk * 4 + half * 2;
    v2f a, b;
    a.x = A[(ti * 16 + l) * LDW + k0];
    a.y = A[(ti * 16 + l) * LDW + k0 + 1];
    b.x = B[(k0)     * LDW + tj * 16 + l];
    b.y = B[(k0 + 1) * LDW + tj * 16 + l];
    acc = __builtin_amdgcn_wmma_f32_16x16x4_f32(false, a, false, b,
                                                (short)0, acc, false, false);
  }
#pragma unroll
  for (int v = 0; v < 8; ++v) {
    const int row = ti * 16 + half * 8 + v;
    D[row * LDW + tj * 16 + l] = acc[v];
  }
}

__global__ __launch_bounds__(128) void logeig32_kernel(
    const float* __restrict__ P, float* __restrict__ out, int nmat) {
  __shared__ float Ys[32 * LDW];
  __shared__ float Zs[32 * LDW];
  __shared__ float Ws[32 * LDW];
  __shared__ float NYs[32 * LDW];
  __shared__ float NZs[32 * LDW];
  __shared__ float red[34];

  const int mat = blockIdx.x;
  if (mat >= nmat) return;  // block-uniform: EXEC stays all-ones for WMMA

  const int tid  = threadIdx.x;
  const int lane = tid & 31;
  const int wave = tid >> 5;
  const int ti   = wave >> 1;
  const int tj   = wave & 1;
  const size_t base = (size_t)mat * 1024;

  // ---- TDM: DMA the 32x32 f32 tile into Ws with hardware 33-float pitch --
  if (wave == 0) {
    const unsigned long long ga = (unsigned long long)(P + base);
    const unsigned int lds = (unsigned int)(uintptr_t)Ws;  // LDS byte offset

    v4u g0;
    g0[0] = 1u;                                   // count=1, user mode
    g0[1] = lds;                                  // lds_addr
    g0[2] = (unsigned int)(ga & 0xFFFFFFFFu);     // global_addr[31:0]
    g0[3] = (unsigned int)((ga >> 32) & 0x01FFFFFFu) | (2u << 30);  // type=2

    v8i g1;
    g1[0] = (int)((2u << 16)      // data_size = 4 bytes
                 | (1u << 20)     // pad_enable
                 | (4u << 22));   // pad_interval: every 32 DWORDs
                                  // pad_amount = 0 -> 1 DWORD pad (pitch 33)
    g1[1] = (int)(32u << 16);     // tensor_dim0 = 32 (low 16 bits at bit 48)
    g1[2] = (int)(32u << 16);     // tensor_dim1 = 32 (low 16 bits at bit 80)
    g1[3] = (int)(32u << 16);     // tile_dim0   = 32 (bits 127:112)
    g1[4] = 32;                   // tile_dim1   = 32 (bits 143:128)
    g1[5] = 32;                   // tensor_dim0_stride = 32 (bits 207:160)
    g1[6] = 0;
    g1[7] = 0;

    v4i gz4 = {0, 0, 0, 0};               // groups 2/3 unused (2D tensor)
    v8i gz8 = {0, 0, 0, 0, 0, 0, 0, 0};   // extra group (clang-23 form)

    // 6-arg form: (g0:v4u, g1:v8i, g2:v4i, g3:v4i, g4:v8i, cpol:i32)
    __builtin_amdgcn_tensor_load_to_lds(g0, g1, gz4, gz4, gz8, 0);
    __builtin_amdgcn_s_wait_tensorcnt(0);
  }
  __syncthreads();

  // ---- trace reduction ---------------------------------------------------
  if (tid < 32) red[tid] = Ws[tid * LDW + tid];
  __syncthreads();
  if (tid == 0) {
    float s = 0.f;
#pragma unroll
    for (int i = 0; i < 32; ++i) s += red[i];
    red[32] = s;
  }
  __syncthreads();
  const float trc  = red[32];     // trace(P) >= 32*eps > 0 (SPD)
  const float invc = 1.0f / trc;
  const float logc = logf(trc);

  // ---- Y = sym(P)/c ; Z = I ---------------------------------------------
#pragma unroll
  for (int i = 0; i < 8; ++i) {
    const int e = tid + i * 128;
    const int r = e >> 5, c = e & 31;
    Ys[r * LDW + c] = 0.5f * (Ws[r * LDW + c] + Ws[c * LDW + r]) * invc;
    Zs[r * LDW + c] = (r == c) ? 1.0f : 0.0f;
  }
  __syncthreads();

  // ---- m = 6 chained square roots, coupled Newton-Schulz, fused ----------
  constexpr int M_LVL = 6;
  for (int lvl = 0; lvl < M_LVL; ++lvl) {
    const int pairs = (lvl == 0) ? 6 : 4;  // 12 / 8 iterations (even)
    for (int p = 0; p < pairs; ++p) {
      // iteration 1: (Y,Z) -> (NY,NZ)
      mm32s(Zs, Ys, Ws, -0.5f, lane, ti, tj);          // W = -0.5*Z*Y
      __syncthreads();
      mm32c(Ys, Ws, Ys, 1.5f, NYs, lane, ti, tj);      // NY = Y*W + 1.5*Y
      mm32c(Ws, Zs, Zs, 1.5f, NZs, lane, ti, tj);      // NZ = W*Z + 1.5*Z
      __syncthreads();
      // iteration 2: (NY,NZ) -> (Y,Z)
      mm32s(NZs, NYs, Ws, -0.5f, lane, ti, tj);
      __syncthreads();
      mm32c(NYs, Ws, NYs, 1.5f, Ys, lane, ti, tj);
      mm32c(Ws, NZs, NZs, 1.5f, Zs, lane, ti, tj);
      __syncthreads();
    }
    // next level: A <- Y (already in Ys), Z <- I
#pragma unroll
    for (int i = 0; i < 8; ++i) {
      const int e = tid + i * 128;
      const int r = e >> 5, c = e & 31;
      Zs[r * LDW + c] = (r == c) ? 1.0f : 0.0f;
    }
    __syncthreads();
  }

  // ---- log series: E = Y - I ; log(A) ~= 2^m (E - E^2/2 + E^3/3 - E^4/4) -
#pragma unroll
  for (int i = 0; i < 8; ++i) {
    const int e = tid + i * 128;
    const int r = e >> 5, c = e & 31;
    Ws[r * LDW + c] = Ys[r * LDW + c] - ((r == c) ? 1.0f : 0.0f);
  }
  __syncthreads();
  mm32s(Ws, Ws, NYs, 1.0f, lane, ti, tj);   // E^2
  __syncthreads();
  mm32s(NYs, Ws, NZs, 1.0f, lane, ti, tj);  // E^3
  mm32s(NYs, NYs, Zs, 1.0f, lane, ti, tj);  // E^4 (Z buffer is free now)
  __syncthreads();

  const float scale = 64.0f;                // 2^M_LVL
#pragma unroll
  for (int i = 0; i < 8; ++i) {
    const int e = tid + i * 128;
    const int r = e >> 5, c = e & 31;
    const int idx = r * LDW + c;
    float v = scale * (Ws[idx] - 0.5f * NYs[idx] +
                       (1.0f / 3.0f) * NZs[idx] - 0.25f * Zs[idx]);
    if (r == c) v += logc;
    out[base + e] = v;
  }
}

extern "C" void kernel_launch(void* const* d_in, const int* in_sizes, int n_in,
                              void* d_out, int out_size, void* d_ws, size_t ws_size,
                              hipStream_t stream) {
  const float* P = (const float*)d_in[0];
  float* out = (float*)d_out;
  const int nmat = in_sizes[0] / 1024;  // (B*H) matrices of 32x32
  logeig32_kernel<<<nmat, 128, 0, stream>>>(P, out, nmat);
}